// HungarianMatcher_67886253080772
// MI455X (gfx1250) — compile-verified
//
#include <hip/hip_runtime.h>
#include <stdint.h>

// Problem constants (fixed by the reference)
#define BSZ    16
#define QN     1024
#define KCLS   256
#define TN     128
#define NROWS  (BSZ * QN)      // 16384 predictions
#define MCOLS  (BSZ * TN)      // 2048 targets
#define RPB    16              // rows per block
#define THREADS 256
#define MPT    (MCOLS / THREADS) // 8 columns per thread

#define AS1 __attribute__((address_space(1)))
#define AS3 __attribute__((address_space(3)))

typedef int v2i __attribute__((ext_vector_type(2)));

// ---- CDNA5 async global->LDS copies (ASYNCcnt path) -----------------------
__device__ __forceinline__ void async_ld_b32(const void* g, void* l) {
#if __has_builtin(__builtin_amdgcn_global_load_async_to_lds_b32)
  __builtin_amdgcn_global_load_async_to_lds_b32(
      (AS1 int*)(uintptr_t)g, (AS3 int*)(AS3 void*)l, 0, 0);
#else
  unsigned loff = (unsigned)(uintptr_t)(AS3 void*)l;
  asm volatile("global_load_async_to_lds_b32 %0, %1, off"
               :: "v"(loff), "v"((uint64_t)(uintptr_t)g) : "memory");
#endif
}

__device__ __forceinline__ void async_ld_b64(const void* g, void* l) {
#if __has_builtin(__builtin_amdgcn_global_load_async_to_lds_b64)
  __builtin_amdgcn_global_load_async_to_lds_b64(
      (AS1 v2i*)(uintptr_t)g, (AS3 v2i*)(AS3 void*)l, 0, 0);
#else
  unsigned loff = (unsigned)(uintptr_t)(AS3 void*)l;
  asm volatile("global_load_async_to_lds_b64 %0, %1, off"
               :: "v"(loff), "v"((uint64_t)(uintptr_t)g) : "memory");
#endif
}

__device__ __forceinline__ void wait_async0() {
#if __has_builtin(__builtin_amdgcn_s_wait_asynccnt)
  __builtin_amdgcn_s_wait_asynccnt(0);
#else
  asm volatile("s_wait_asynccnt 0" ::: "memory");
#endif
}

// ---------------------------------------------------------------------------
__global__ __launch_bounds__(THREADS)
void matcher_cost_kernel(const float* __restrict__ logits,   // [16,1024,256]
                         const float* __restrict__ pboxes,   // [16,1024,2] (mid,w)
                         const float* __restrict__ tboxes,   // [16,128,2]  (x0,x1)
                         const int*   __restrict__ tlabels,  // [16,128]
                         float* __restrict__ C)               // [16,1024,2048]
{
  __shared__ float2 s_tbox[MCOLS];        // 16 KB
  __shared__ int    s_tid [MCOLS];        //  8 KB
  __shared__ float  s_prob[RPB * KCLS];   // 16 KB
  __shared__ float  s_px0[RPB], s_px1[RPB], s_pw[RPB];

  const int tid  = threadIdx.x;
  const int row0 = blockIdx.x * RPB;

  // ---- Phase A: async-stage target boxes + labels into LDS (overlapped) ---
  #pragma unroll
  for (int i = 0; i < MPT; ++i) {
    const int m = i * THREADS + tid;
    async_ld_b64(&((const float2*)tboxes)[m], &s_tbox[m]);
    async_ld_b32(&tlabels[m],                &s_tid[m]);
  }

  // pred boxes (mid,w) -> (x0,x1,area) for this block's 16 rows
  if (tid < RPB) {
    const float2 pb = ((const float2*)pboxes)[row0 + tid];
    s_px0[tid] = pb.x - 0.5f * pb.y;
    s_px1[tid] = pb.x + 0.5f * pb.y;
    s_pw [tid] = pb.y;
  }

  // ---- Phase B: softmax, one row per wave per pass (8 waves x 2 rows) ---------
  const int wave = tid >> 5, lane = tid & 31;
  #pragma unroll
  for (int rr = 0; rr < 2; ++rr) {
    const int r = wave * 2 + rr;
    const float* lrow = logits + (size_t)(row0 + r) * KCLS;
    float l[8];
    #pragma unroll
    for (int j = 0; j < 8; ++j) l[j] = lrow[lane + 32 * j];

    float mx = l[0];
    #pragma unroll
    for (int j = 1; j < 8; ++j) mx = fmaxf(mx, l[j]);
    #pragma unroll
    for (int off = 16; off > 0; off >>= 1) mx = fmaxf(mx, __shfl_xor(mx, off, 32));

    float e[8], s = 0.0f;
    #pragma unroll
    for (int j = 0; j < 8; ++j) {
      e[j] = exp2f((l[j] - mx) * 1.44269504088896340736f);  // v_exp_f32
      s += e[j];
    }
    #pragma unroll
    for (int off = 16; off > 0; off >>= 1) s += __shfl_xor(s, off, 32);

    const float inv = __builtin_amdgcn_rcpf(s);
    #pragma unroll
    for (int j = 0; j < 8; ++j) s_prob[r * KCLS + lane + 32 * j] = e[j] * inv;
  }

  wait_async0();     // our async LDS writes landed
  __syncthreads();   // everyone's LDS (prob rows + tgt stage) visible

  // ---- Phase C: cost matrix, 8 columns/thread held in VGPRs ---------------
  float tx0[MPT], tx1[MPT], ta[MPT];
  int   cls[MPT];
  #pragma unroll
  for (int i = 0; i < MPT; ++i) {
    const int m = i * THREADS + tid;
    const float2 tb = s_tbox[m];
    tx0[i] = tb.x; tx1[i] = tb.y; ta[i] = tb.y - tb.x;
    cls[i] = s_tid[m];
  }

  float* outbase = C + (size_t)row0 * MCOLS + tid;
  for (int r = 0; r < RPB; ++r) {
    const float px0 = s_px0[r], px1 = s_px1[r], pw = s_pw[r];
    const float* pr = &s_prob[r * KCLS];
    float* orow = outbase + (size_t)r * MCOLS;
    #pragma unroll
    for (int i = 0; i < MPT; ++i) {
      const float p     = pr[cls[i]];                       // LDS gather
      const float db    = fabsf(px0 - tx0[i]) + fabsf(px1 - tx1[i]);
      const float lt    = fmaxf(px0, tx0[i]);
      const float rb    = fminf(px1, tx1[i]);
      const float inter = fmaxf(rb - lt, 0.0f);
      const float uni   = pw + ta[i] - inter;
      const float hull  = fmaxf(px1, tx1[i]) - fminf(px0, tx0[i]);
      const float iou   = inter * __builtin_amdgcn_rcpf(uni);
      const float giou  = iou - (hull - uni) * __builtin_amdgcn_rcpf(hull);
      // C = 5*bbox - p - 2*giou
      const float cost  = fmaf(5.0f, db, fmaf(-2.0f, giou, -p));
      __builtin_nontemporal_store(cost, orow + i * THREADS);
    }
  }
}

// ---------------------------------------------------------------------------
extern "C" void kernel_launch(void* const* d_in, const int* in_sizes, int n_in,
                              void* d_out, int out_size, void* d_ws, size_t ws_size,
                              hipStream_t stream) {
  const float* logits  = (const float*)d_in[0];  // pred_logits
  const float* pboxes  = (const float*)d_in[1];  // pred_boxes
  const float* tboxes  = (const float*)d_in[2];  // tgt_boxes
  const int*   tlabels = (const int*)  d_in[3];  // tgt_labels
  float* C = (float*)d_out;

  dim3 grid(NROWS / RPB);   // 1024 blocks
  dim3 block(THREADS);      // 256 threads = 8 wave32
  matcher_cost_kernel<<<grid, block, 0, stream>>>(logits, pboxes, tboxes, tlabels, C);
}